// PrecRec_83494164234585
// MI455X (gfx1250) — compile-verified
//
#include <hip/hip_runtime.h>
#include <hip/hip_bf16.h>

typedef float v2f __attribute__((ext_vector_type(2)));
typedef float v8f __attribute__((ext_vector_type(8)));

#define NT 10              // number of thresholds
#define NBINS (NT + 1)     // 11 histogram bins (j = 0..10)

// Bin one element: j = #{ thr_i < sigmoid(p) } = clamp(floor(12*sigmoid(p)), 0, 10).
// Pack (target&mask) count in high 32 bits, mask count in low 32 bits, and do a
// single packed LDS u64 atomic add -> ds_add_u64.
__device__ __forceinline__ void acc_one(float p, int m, int t,
                                        unsigned long long* hist) {
  float ex = __expf(-p);                        // v_mul + v_exp_f32
  float s  = __builtin_amdgcn_rcpf(1.0f + ex);  // v_add + v_rcp_f32  (sigmoid)
  int j = (int)(12.0f * s);                     // v_mul + v_cvt (floor, s >= 0)
  j = j > NT ? NT : j;                          // v_min
  unsigned mm = (m != 0) ? 1u : 0u;
  unsigned tt = ((t != 0) ? 1u : 0u) & mm;
  unsigned long long inc =
      ((unsigned long long)tt << 32) | (unsigned long long)mm;
  atomicAdd(&hist[j], inc);                     // ds_add_u64 (packed dual counter)
}

__global__ void precrec_zero(unsigned* gws) {
  if (threadIdx.x < 2 * NBINS) gws[threadIdx.x] = 0u;
}

__global__ void precrec_hist(const float* __restrict__ pred,
                             const int* __restrict__ mask,
                             const int* __restrict__ tgt,
                             unsigned* __restrict__ gws,
                             long long n) {
  __shared__ unsigned long long hist[NBINS];   // 88 B per block
  if (threadIdx.x < NBINS) hist[threadIdx.x] = 0ull;
  __syncthreads();

  long long nvec   = n >> 2;
  long long i0     = (long long)blockIdx.x * blockDim.x + threadIdx.x;
  long long stride = (long long)gridDim.x * blockDim.x;

  const float4* p4 = (const float4*)pred;
  const int4*   m4 = (const int4*)mask;
  const int4*   t4 = (const int4*)tgt;

  for (long long v = i0; v < nvec; v += stride) {   // b128 streaming loads
    float4 p = p4[v];
    int4   m = m4[v];
    int4   t = t4[v];
    acc_one(p.x, m.x, t.x, hist);
    acc_one(p.y, m.y, t.y, hist);
    acc_one(p.z, m.z, t.z, hist);
    acc_one(p.w, m.w, t.w, hist);
  }
  for (long long i = (nvec << 2) + i0; i < n; i += stride)  // scalar tail
    acc_one(pred[i], mask[i], tgt[i], hist);

  __syncthreads();
  // 22 u32 words: word 2j = mask-count(bin j), word 2j+1 = target-count(bin j)
  if (threadIdx.x < 2 * NBINS)
    atomicAdd(&gws[threadIdx.x], ((const unsigned*)hist)[threadIdx.x]);
}

// Finalize with one wave. tp_i / poscnt_i are suffix sums of the histograms;
// [tp, fp, tn, fn] = A(4x4) * [tp, poscnt, n_tgt, n_mask - n_tgt] per threshold
// column -> exact f32 V_WMMA_F32_16X16X4_F32 (all counts < 2^24).
__global__ void precrec_final(const unsigned* __restrict__ gws,
                              float* __restrict__ out) {
  int  lane = threadIdx.x;     // launched with exactly 32 threads, EXEC all-1s
  int  n    = lane & 15;
  bool hi   = lane >= 16;

  float tp = 0.f, pc = 0.f, ntg = 0.f, nmk = 0.f;
#pragma unroll
  for (int j = 0; j < NBINS; ++j) {
    float hm = (float)gws[2 * j];
    float ht = (float)gws[2 * j + 1];
    nmk += hm;
    ntg += ht;
    if (j > n) { pc += hm; tp += ht; }   // pos_i  <=>  bin j > i
  }
  float nrest = nmk - ntg;

  // B (4x16): VGPR0 = {K=0 | K=2}, VGPR1 = {K=1 | K=3} across lane halves.
  v2f B;
  B.x = hi ? ntg   : (n < NT ? tp : 0.f);
  B.y = hi ? nrest : (n < NT ? pc : 0.f);

  // A (16x4): rows M = [tp, fp, tn, fn] transform; VGPR0 = {K=0 | K=2}, VGPR1 = {K=1 | K=3}.
  //   tp =  tp
  //   fp = -tp + poscnt
  //   tn =  tp - poscnt + (n_mask - n_tgt)
  //   fn = -tp + n_tgt
  const float T[4][4] = {{ 1.f, 0.f, 0.f, 0.f},
                         {-1.f, 1.f, 0.f, 0.f},
                         { 1.f,-1.f, 0.f, 1.f},
                         {-1.f, 0.f, 1.f, 0.f}};
  int k0 = hi ? 2 : 0, k1 = hi ? 3 : 1;
  v2f A;
  A.x = (n < 4) ? T[n][k0] : 0.f;
  A.y = (n < 4) ? T[n][k1] : 0.f;

  v8f C = {0.f, 0.f, 0.f, 0.f, 0.f, 0.f, 0.f, 0.f};
  v8f D = __builtin_amdgcn_wmma_f32_16x16x4_f32(false, A, false, B,
                                                (short)0, C, false, false);

  // D VGPR r on lanes 0..15 holds row M=r, col N=lane. Rows 0..3, cols 0..9.
  if (lane < NT) {
    out[0 * NT + lane] = D[0];
    out[1 * NT + lane] = D[1];
    out[2 * NT + lane] = D[2];
    out[3 * NT + lane] = D[3];
  }
}

extern "C" void kernel_launch(void* const* d_in, const int* in_sizes, int n_in,
                              void* d_out, int out_size, void* d_ws, size_t ws_size,
                              hipStream_t stream) {
  const float* pred = (const float*)d_in[0];
  const int*   mask = (const int*)d_in[1];
  const int*   tgt  = (const int*)d_in[2];
  unsigned* gws = (unsigned*)d_ws;   // 22 u32 counters
  float*    out = (float*)d_out;     // 40 floats: rows [tp, fp, tn, fn] x 10 thresholds
  long long n = in_sizes[0];

  precrec_zero<<<1, 32, 0, stream>>>(gws);

  long long nvec = n >> 2;
  int threads = 256;                                   // 8 wave32s per block
  long long want = (nvec + threads - 1) / threads;
  int blocks = (int)(want < 1024 ? (want > 0 ? want : 1) : 1024);
  precrec_hist<<<blocks, threads, 0, stream>>>(pred, mask, tgt, gws, n);

  precrec_final<<<1, 32, 0, stream>>>(gws, out);
}